// GNNLayer_21457656611215
// MI455X (gfx1250) — compile-verified
//
#include <hip/hip_runtime.h>

typedef float v2f __attribute__((ext_vector_type(2)));
typedef float v8f __attribute__((ext_vector_type(8)));

#define DIM 128

__device__ __forceinline__ float gelu_exact(float x) {
    // exact GELU: 0.5*x*(1+erf(x/sqrt(2))) — matches torch nn.GELU default
    return 0.5f * x * (1.0f + erff(x * 0.70710678118654752440f));
}

// -------------------------------------------------------------------------
// Kernel 1: node = x @ Wn^T + bn            [N,128] = [N,128]x[128,128]
// Block = 256 threads (8 waves). Block computes a 16-row stripe; wave w
// computes the 16x16 output tile at columns [16w,16w+16).
// WMMA f32 16x16x4, K-loop of 32 steps, fp32 accumulate (matches reference).
// -------------------------------------------------------------------------
__global__ __launch_bounds__(256) void k_node_linear(
    const float* __restrict__ x, const float* __restrict__ Wn,
    const float* __restrict__ bn, float* __restrict__ node)
{
    const int tid  = threadIdx.x;
    const int wave = tid >> 5;
    const int lane = tid & 31;
    const int m    = lane & 15;       // A-matrix row / B column within tile
    const int half = lane >> 4;       // which K-half of the fragment
    const int kh   = half * 2;
    const int ncol = wave * 16 + m;   // output column
    const int row0 = blockIdx.x * 16; // output row base (N % 16 == 0)

    const float* arow = x  + (size_t)(row0 + m) * DIM + kh;
    const float* brow = Wn + (size_t)ncol * DIM + kh;   // B[k][n] = Wn[n][k]

    v8f c = {};
    #pragma unroll 4
    for (int k0 = 0; k0 < DIM; k0 += 4) {
        v2f a = *(const v2f*)(arow + k0);
        v2f b = *(const v2f*)(brow + k0);
        c = __builtin_amdgcn_wmma_f32_16x16x4_f32(false, a, false, b,
                                                  (short)0, c, false, false);
    }
    const float bias = bn[ncol];
    #pragma unroll
    for (int v = 0; v < 8; ++v) {
        int row = row0 + v + half * 8;          // C/D layout: VGPR v -> M = v (+8)
        node[(size_t)row * DIM + ncol] = c[v] + bias;
    }
}

// -------------------------------------------------------------------------
// Kernel 2 (fused edge pass): per 16-edge tile,
//   * async-DMA gather node[src]|node[dst] -> LDS [16 x 256]
//     (GLOBAL_LOAD_ASYNC_TO_LDS_B128: per-lane global addr -> per-lane LDS
//      addr, tracked by ASYNCcnt — no VGPR round trip for the 820 MB gather)
//   * atomicAdd node[dst] into nbr_sum[src]     (segment_sum #1)
//   * edge_emb = tile @ We^T + be  via WMMA, K=256
//   * atomicAdd edge_emb into edge_sum[src]     (segment_sum #2)
// node / accumulators are L2-resident (25.6 MB each vs 192 MB L2), so the
// random gathers and atomics stay on-chip.
// -------------------------------------------------------------------------
__global__ __launch_bounds__(256) void k_edge(
    const float* __restrict__ node, const long long* __restrict__ eidx,
    const float* __restrict__ We, const float* __restrict__ be,
    float* __restrict__ nbr_sum, float* __restrict__ edge_sum, long long E)
{
    __shared__ float sIn[16 * 256];
    __shared__ int   sSrc[16];

    const int tid = threadIdx.x;
    const long long e0 = (long long)blockIdx.x * 16;   // E % 16 == 0
    const long long* srcIdx = eidx;        // edge_index[0, :]
    const long long* dstIdx = eidx + E;    // edge_index[1, :]

    if (tid < 16) sSrc[tid] = (int)srcIdx[e0 + tid];

    // Gather 16 x 256 floats as 1024 float4 chunks (4 per thread), DMA'd
    // straight into LDS. Both sides are 16B-aligned.
    #pragma unroll
    for (int it = 0; it < 4; ++it) {
        int i   = tid + it * 256;
        int row = i >> 6;                 // edge within tile
        int c4  = i & 63;                 // float4 index within 256 floats
        long long nid = (c4 < 32) ? srcIdx[e0 + row] : dstIdx[e0 + row];
        int col = (c4 & 31) * 4;          // 0..124 within the node row
        const float* gsrc = node + nid * DIM + col;
        unsigned lds_addr = (unsigned)(unsigned long long)(const void*)
                            (&sIn[row * 256 + c4 * 4]);
        asm volatile("global_load_async_to_lds_b128 %0, %1, off"
                     :: "v"(lds_addr), "v"(gsrc) : "memory");
    }
    asm volatile("s_wait_asynccnt 0x0" ::: "memory");
    __syncthreads();

    // Scatter-add node[dst] (cols 128..255 of the staged tile) into nbr_sum[src].
    for (int i = tid; i < 16 * 128; i += 256) {
        int row = i >> 7;
        int col = i & 127;
        atomicAdd(&nbr_sum[(size_t)sSrc[row] * DIM + col],
                  sIn[row * 256 + 128 + col]);
    }

    // Edge GEMM: [16 x 256] @ We^T -> [16 x 128]; wave w owns cols 16w..16w+15.
    const int wave = tid >> 5;
    const int lane = tid & 31;
    const int m    = lane & 15;
    const int half = lane >> 4;
    const int kh   = half * 2;
    const int ncol = wave * 16 + m;
    const float* brow = We + (size_t)ncol * 256 + kh;  // B[k][n] = We[n][k]

    v8f c = {};
    #pragma unroll 4
    for (int k0 = 0; k0 < 256; k0 += 4) {
        v2f a = *(const v2f*)(&sIn[m * 256 + k0 + kh]);
        v2f b = *(const v2f*)(brow + k0);
        c = __builtin_amdgcn_wmma_f32_16x16x4_f32(false, a, false, b,
                                                  (short)0, c, false, false);
    }
    const float bias = be[ncol];
    #pragma unroll
    for (int v = 0; v < 8; ++v) {
        int row = v + half * 8;           // edge row within tile
        atomicAdd(&edge_sum[(size_t)sSrc[row] * DIM + ncol], c[v] + bias);
    }
}

// -------------------------------------------------------------------------
// Kernel 3: out = gelu([node | nbr_sum | edge_sum]) @ Wu^T + bu
// GELU applied while staging the [16 x 384] tile into LDS; K=384 WMMA loop.
// -------------------------------------------------------------------------
__global__ __launch_bounds__(256) void k_out(
    const float* __restrict__ node, const float* __restrict__ nbr,
    const float* __restrict__ esum, const float* __restrict__ Wu,
    const float* __restrict__ bu, float* __restrict__ out)
{
    __shared__ float sP[16 * 384];
    const int tid  = threadIdx.x;
    const int row0 = blockIdx.x * 16;

    for (int i = tid; i < 16 * 384; i += 256) {
        int row = i / 384;
        int col = i - row * 384;
        size_t base = (size_t)(row0 + row) * DIM;
        float val = (col < 128) ? node[base + col]
                  : (col < 256) ? nbr[base + (col - 128)]
                                : esum[base + (col - 256)];
        sP[row * 384 + col] = gelu_exact(val);
    }
    __syncthreads();

    const int wave = tid >> 5;
    const int lane = tid & 31;
    const int m    = lane & 15;
    const int half = lane >> 4;
    const int kh   = half * 2;
    const int ncol = wave * 16 + m;
    const float* brow = Wu + (size_t)ncol * 384 + kh;  // B[k][n] = Wu[n][k]

    v8f c = {};
    #pragma unroll 4
    for (int k0 = 0; k0 < 384; k0 += 4) {
        v2f a = *(const v2f*)(&sP[m * 384 + k0 + kh]);
        v2f b = *(const v2f*)(brow + k0);
        c = __builtin_amdgcn_wmma_f32_16x16x4_f32(false, a, false, b,
                                                  (short)0, c, false, false);
    }
    const float bias = bu[ncol];
    #pragma unroll
    for (int v = 0; v < 8; ++v) {
        int row = row0 + v + half * 8;
        out[(size_t)row * DIM + ncol] = c[v] + bias;
    }
}

// -------------------------------------------------------------------------
// Host-side launcher. Inputs: x, edge_index(int64), Wn, bn, We, be, Wu, bu.
// Workspace layout: node [N*128] | nbr_sum [N*128] | edge_sum [N*128] (f32).
// -------------------------------------------------------------------------
extern "C" void kernel_launch(void* const* d_in, const int* in_sizes, int n_in,
                              void* d_out, int out_size, void* d_ws, size_t ws_size,
                              hipStream_t stream) {
    const float*     x    = (const float*)d_in[0];
    const long long* eidx = (const long long*)d_in[1];   // int64 [2,E]
    const float*     Wn   = (const float*)d_in[2];
    const float*     bn   = (const float*)d_in[3];
    const float*     We   = (const float*)d_in[4];
    const float*     be   = (const float*)d_in[5];
    const float*     Wu   = (const float*)d_in[6];
    const float*     bu   = (const float*)d_in[7];
    float*           out  = (float*)d_out;

    const int       N = in_sizes[0] / DIM;       // 50000 (divisible by 16)
    const long long E = in_sizes[1] / 2;         // 800000 (divisible by 16)

    float* node = (float*)d_ws;
    float* nbr  = node + (size_t)N * DIM;
    float* esum = nbr  + (size_t)N * DIM;

    // Zero the two segment-sum accumulators every call (capture-safe).
    hipMemsetAsync(nbr, 0, (size_t)2 * N * DIM * sizeof(float), stream);

    k_node_linear<<<N / 16, 256, 0, stream>>>(x, Wn, bn, node);
    k_edge<<<(int)(E / 16), 256, 0, stream>>>(node, eidx, We, be, nbr, esum, E);
    k_out<<<N / 16, 256, 0, stream>>>(node, nbr, esum, Wu, bu, out);
}